// LinearEquivariant1to2_28870770163777
// MI455X (gfx1250) — compile-verified
//
#include <hip/hip_runtime.h>

typedef __attribute__((ext_vector_type(2))) float v2f;
typedef __attribute__((ext_vector_type(8))) float v8f;

#define B_DIM 8
#define I_DIM 16
#define O_DIM 16
#define D_DIM 512   // 2^9

// ---------------------------------------------------------------------------
// Kernel 1: tiny precompute. One wave (32 lanes) per batch b.
//   rowv[b,o,d]  = y2  (row-broadcast term)
//   colv[b,o,d]  = y3  (col-broadcast term)
//   diagv[b,o,d] = y1 + y4[b,o]  (added only on the diagonal)
//   cconst[b,o]  = y5 + bias[o]
// y1/y2/y3 computed with exact-f32 V_WMMA_F32_16X16X4_F32 (K=16 -> 4 steps).
// ---------------------------------------------------------------------------
__global__ __launch_bounds__(32)
void equiv_precompute(const float* __restrict__ x,      // (B, I, D)
                      const float* __restrict__ w,      // (O, 5*I)
                      const float* __restrict__ bias,   // (O)
                      float* __restrict__ rowv,
                      float* __restrict__ colv,
                      float* __restrict__ diagv,
                      float* __restrict__ cconst)
{
  const int b    = blockIdx.x;          // 0..7
  const int lane = threadIdx.x;         // 0..31
  const int m    = lane & 15;           // lane-within-half (M index / N index)
  const int half = lane >> 4;           // 0 or 1

  const float* __restrict__ xb = x + (size_t)b * I_DIM * D_DIM;

  __shared__ float s_sh[16];
  __shared__ float y4_sh[16];

  // ---- s[i] = sum_d x[b,i,d] : lane handles i=m, one half of D ----
  float partial = 0.f;
  {
    const float* xi = xb + m * D_DIM + half * (D_DIM / 2);
    #pragma unroll 8
    for (int d = 0; d < D_DIM / 2; ++d) partial += xi[d];
  }
  partial += __shfl_xor(partial, 16, 32);   // combine the two halves
  if (lane < 16) s_sh[lane] = partial;
  __syncthreads();

  // ---- y4[o], y5[o] : lanes 0..15 each own o = lane ----
  {
    float y4 = 0.f, y5 = 0.f;
    #pragma unroll
    for (int i = 0; i < 16; ++i) {
      const float si = s_sh[i];
      y4 += w[m * 80 + 3 * 16 + i] * si;   // basis op4 (sum on diagonal)
      y5 += w[m * 80 + 4 * 16 + i] * si;   // basis op5 (sum everywhere)
    }
    if (lane < 16) {
      y4_sh[lane] = y4;
      cconst[b * O_DIM + lane] = y5 + bias[lane];
    }
  }
  __syncthreads();

  // ---- A fragments for terms k=0 (y1 diag), 1 (y2 row), 2 (y3 col) ----
  // 16x4 f32 A layout: lanes 0-15 -> K=(kk*4+0, kk*4+1); lanes 16-31 -> (+2,+3)
  v2f afrag[3][4];
  #pragma unroll
  for (int k = 0; k < 3; ++k) {
    #pragma unroll
    for (int kk = 0; kk < 4; ++kk) {
      const int kbase = kk * 4 + half * 2;
      afrag[k][kk][0] = w[m * 80 + k * 16 + kbase + 0];
      afrag[k][kk][1] = w[m * 80 + k * 16 + kbase + 1];
    }
  }

  // ---- y1/y2/y3 = Wk (16x16) @ x[b] (16x512), 16-column tiles ----
  for (int dt = 0; dt < D_DIM / 16; ++dt) {
    v8f acc0 = {}; v8f acc1 = {}; v8f acc2 = {};
    #pragma unroll
    for (int kk = 0; kk < 4; ++kk) {
      // 4x16 f32 B layout: v0: lanes0-15=K row (kk*4+half*2), v1: next row
      const int rowb = kk * 4 + half * 2;
      v2f bfrag;
      bfrag[0] = xb[(rowb + 0) * D_DIM + dt * 16 + m];
      bfrag[1] = xb[(rowb + 1) * D_DIM + dt * 16 + m];
      acc0 = __builtin_amdgcn_wmma_f32_16x16x4_f32(false, afrag[0][kk], false, bfrag,
                                                   (short)0, acc0, false, false);
      acc1 = __builtin_amdgcn_wmma_f32_16x16x4_f32(false, afrag[1][kk], false, bfrag,
                                                   (short)0, acc1, false, false);
      acc2 = __builtin_amdgcn_wmma_f32_16x16x4_f32(false, afrag[2][kk], false, bfrag,
                                                   (short)0, acc2, false, false);
    }
    // C/D 16x16 f32 layout: vgpr j, lanes 0-15 -> M=j, lanes 16-31 -> M=j+8
    const int d = dt * 16 + m;
    #pragma unroll
    for (int j = 0; j < 8; ++j) {
      const int o = half * 8 + j;
      const size_t idx = ((size_t)b * O_DIM + o) * D_DIM + d;
      diagv[idx] = acc0[j] + y4_sh[o];
      rowv[idx]  = acc1[j];
      colv[idx]  = acc2[j];
    }
  }
}

// ---------------------------------------------------------------------------
// Kernel 2: bandwidth-bound streamer (no diagonal logic at all).
//   out[b,o,r,c] = rowv[bo,r] + cconst[bo] + colv[bo,c]
// Each thread emits 16 consecutive columns of one row: 4x global_store_b128.
// No branches, no dynamic indexing -> minimal instructions per stored byte.
// ---------------------------------------------------------------------------
__global__ __launch_bounds__(256)
void equiv_write(const float* __restrict__ rowv,
                 const float* __restrict__ colv,
                 const float* __restrict__ cconst,
                 float* __restrict__ out)
{
  const size_t tid  = (size_t)blockIdx.x * 256 + threadIdx.x;
  const size_t base = tid * 16;                      // flat idx into (B*O, D, D)
  const int c0 = (int)(base & (D_DIM - 1));          // 16-aligned column start
  const int r  = (int)((base >> 9) & (D_DIM - 1));
  const int bo = (int)(base >> 18);

  const float rv = rowv[(size_t)bo * D_DIM + r] + cconst[bo];
  const float4* __restrict__ cp =
      reinterpret_cast<const float4*>(colv + (size_t)bo * D_DIM + c0);
  float4* __restrict__ op = reinterpret_cast<float4*>(out + base);

  #pragma unroll
  for (int q = 0; q < 4; ++q) {
    const float4 cv = cp[q];
    op[q] = make_float4(rv + cv.x, rv + cv.y, rv + cv.z, rv + cv.w);
  }
}

// ---------------------------------------------------------------------------
// Kernel 3: diagonal fixup. 65,536 elements (1/512 of output), RMW in L2.
//   out[bo, r, r] += diagv[bo, r]
// ---------------------------------------------------------------------------
__global__ __launch_bounds__(256)
void equiv_diag_fix(const float* __restrict__ diagv, float* __restrict__ out)
{
  const int idx = blockIdx.x * 256 + threadIdx.x;    // idx = bo*512 + r
  const int r = idx & (D_DIM - 1);
  const size_t pos = ((size_t)idx << 9) + (size_t)r; // (bo*512 + r)*512 + r
  out[pos] += diagv[idx];
}

extern "C" void kernel_launch(void* const* d_in, const int* in_sizes, int n_in,
                              void* d_out, int out_size, void* d_ws, size_t ws_size,
                              hipStream_t stream) {
  const float* x    = (const float*)d_in[0];  // (8,16,512)
  const float* w    = (const float*)d_in[1];  // (16,80)
  const float* bias = (const float*)d_in[2];  // (16)
  float* out = (float*)d_out;                 // (8,16,512,512)

  // Workspace layout (f32): rowv | colv | diagv | cconst  = 3*65536 + 128 elems
  float* rowv   = (float*)d_ws;
  float* colv   = rowv + B_DIM * O_DIM * D_DIM;
  float* diagv  = colv + B_DIM * O_DIM * D_DIM;
  float* cconst = diagv + B_DIM * O_DIM * D_DIM;

  equiv_precompute<<<B_DIM, 32, 0, stream>>>(x, w, bias, rowv, colv, diagv, cconst);

  const int total_threads = (B_DIM * O_DIM * D_DIM * D_DIM) / 16;  // 2,097,152
  equiv_write<<<total_threads / 256, 256, 0, stream>>>(rowv, colv, cconst, out);

  const int diag_threads = B_DIM * O_DIM * D_DIM;                  // 65,536
  equiv_diag_fix<<<diag_threads / 256, 256, 0, stream>>>(diagv, out);
}